// GRUEncoder_61340722921594
// MI455X (gfx1250) — compile-verified
//
#include <hip/hip_runtime.h>
#include <math.h>

typedef float v2f __attribute__((ext_vector_type(2)));
typedef float v8f __attribute__((ext_vector_type(8)));

#define BB 64
#define SS 512
#define II 256
#define HH 512

__device__ __forceinline__ float sigm(float x) { return 1.0f / (1.0f + __expf(-x)); }

// One timestep for BOTH directions of one layer.
// grid: (HH/16, 2)  -- x: output column tile within H, y: direction
// block: 128 threads = 4 waves; wave w computes M-tile (rows 16w..16w+15 of B=64)
// Each wave holds 4 f32 WMMA accumulators: r, z, n_x, n_h (16x16 tiles).
//
// WMMA f32 16x16x4 lane layout (ISA 7.12.2):
//   A: lane L, vgpr v -> M = L%16, K = 2*(L/16)+v   (contiguous float2 per lane)
//   B: lane L, vgpr v -> N = L%16, K = 2*(L/16)+v   (W row-major (N,K) -> contiguous float2)
//   C/D: lane L, vgpr v -> N = L%16, M = v + 8*(L/16)
// The lane-dependent K offset (2*lhalf) is folded into the base pointers so the
// K-loop has a uniform compile-time trip count -> unroll by 8, immediate-offset loads.
template <int IN_SZ>
__global__ __launch_bounds__(128) void gru_step_kernel(
    const float* __restrict__ xin,                         // (B,S,IN_SZ) layer input
    const float* __restrict__ wih0, const float* __restrict__ whh0,
    const float* __restrict__ bih0, const float* __restrict__ bhh0,
    const float* __restrict__ wih1, const float* __restrict__ whh1,
    const float* __restrict__ bih1, const float* __restrict__ bhh1,
    float* __restrict__ h0bufs,                            // dir0: 2*B*H double buffer
    float* __restrict__ h1bufs,                            // dir1: 2*B*H double buffer
    float* __restrict__ yout,                              // (B,S,2H)
    int s)                                                 // step index 0..S-1
{
    const int jt   = blockIdx.x;          // column tile within H: 0..31
    const int dir  = blockIdx.y;          // 0=forward, 1=backward
    const int wave = threadIdx.x >> 5;    // 0..3 -> M tile
    const int lane = threadIdx.x & 31;
    const int lhalf = lane >> 4;          // 0/1
    const int lmod  = lane & 15;
    const int kofs  = 2 * lhalf;          // lane-half K offset, folded into bases

    const int t = dir ? (SS - 1 - s) : s;

    const float* wih = dir ? wih1 : wih0;
    const float* whh = dir ? whh1 : whh0;
    const float* bih = dir ? bih1 : bih0;
    const float* bhh = dir ? bhh1 : bhh0;
    float* hbufs = dir ? h1bufs : h0bufs;

    const float* __restrict__ hprev = hbufs + (size_t)(s & 1) * (BB * HH);
    float* __restrict__ hnext = hbufs + (size_t)((s + 1) & 1) * (BB * HH);

    const int row0 = wave * 16;           // M base within B
    const int colg = jt * 16 + lmod;      // output column within H (this lane's N)

    v8f cr  = {0.f, 0.f, 0.f, 0.f, 0.f, 0.f, 0.f, 0.f};
    v8f cz  = cr;
    v8f cnx = cr;
    v8f cnh = cr;

    // ---- Phase 1: input projection  gates += x_t(16xK) * W_ih^T(Kx16) ----
    {
        const float* __restrict__ pa = xin + ((size_t)(row0 + lmod) * SS + (size_t)t) * IN_SZ + kofs;
        const float* __restrict__ pr = wih + (size_t)(0 * HH + colg) * IN_SZ + kofs;
        const float* __restrict__ pz = wih + (size_t)(1 * HH + colg) * IN_SZ + kofs;
        const float* __restrict__ pn = wih + (size_t)(2 * HH + colg) * IN_SZ + kofs;
        #pragma unroll 8
        for (int kb = 0; kb < IN_SZ / 4; ++kb) {
            v2f a  = *(const v2f*)(pa + kb * 4);
            v2f br = *(const v2f*)(pr + kb * 4);
            v2f bz = *(const v2f*)(pz + kb * 4);
            v2f bn = *(const v2f*)(pn + kb * 4);
            cr  = __builtin_amdgcn_wmma_f32_16x16x4_f32(false, a, false, br, (short)0, cr,  false, false);
            cz  = __builtin_amdgcn_wmma_f32_16x16x4_f32(false, a, false, bz, (short)0, cz,  false, false);
            cnx = __builtin_amdgcn_wmma_f32_16x16x4_f32(false, a, false, bn, (short)0, cnx, false, false);
        }
    }

    // ---- Phase 2: hidden projection  gates += h(16xK) * W_hh^T(Kx16) ----
    {
        const float* __restrict__ pa = hprev + (size_t)(row0 + lmod) * HH + kofs;
        const float* __restrict__ pr = whh + (size_t)(0 * HH + colg) * HH + kofs;
        const float* __restrict__ pz = whh + (size_t)(1 * HH + colg) * HH + kofs;
        const float* __restrict__ pn = whh + (size_t)(2 * HH + colg) * HH + kofs;
        #pragma unroll 8
        for (int kb = 0; kb < HH / 4; ++kb) {
            v2f a  = *(const v2f*)(pa + kb * 4);
            v2f br = *(const v2f*)(pr + kb * 4);
            v2f bz = *(const v2f*)(pz + kb * 4);
            v2f bn = *(const v2f*)(pn + kb * 4);
            cr  = __builtin_amdgcn_wmma_f32_16x16x4_f32(false, a, false, br, (short)0, cr,  false, false);
            cz  = __builtin_amdgcn_wmma_f32_16x16x4_f32(false, a, false, bz, (short)0, cz,  false, false);
            cnh = __builtin_amdgcn_wmma_f32_16x16x4_f32(false, a, false, bn, (short)0, cnh, false, false);
        }
    }

    // ---- Epilogue: gate math on the C/D layout (M = v + 8*lhalf, N = lmod) ----
    const float br_b  = bih[0 * HH + colg] + bhh[0 * HH + colg];
    const float bz_b  = bih[1 * HH + colg] + bhh[1 * HH + colg];
    const float bnx_b = bih[2 * HH + colg];
    const float bnh_b = bhh[2 * HH + colg];

    #pragma unroll
    for (int v = 0; v < 8; ++v) {
        const int row = row0 + v + 8 * lhalf;  // global batch row
        const float r = sigm(cr[v] + br_b);
        const float z = sigm(cz[v] + bz_b);
        const float n = tanhf(cnx[v] + bnx_b + r * (cnh[v] + bnh_b));
        const float hp = hprev[(size_t)row * HH + colg];
        const float hnew = (1.0f - z) * n + z * hp;
        hnext[(size_t)row * HH + colg] = hnew;
        yout[((size_t)row * SS + (size_t)t) * (2 * HH) + (size_t)dir * HH + colg] = hnew;
    }
}

__global__ void zero_f32_kernel(float* __restrict__ p, int n) {
    int i = blockIdx.x * blockDim.x + threadIdx.x;
    if (i < n) p[i] = 0.0f;
}

// Final hidden for one layer: dst[0..B*H) = h_fwd, dst[B*H..2BH) = h_bwd
__global__ void copy_hidden_kernel(float* __restrict__ dst,
                                   const float* __restrict__ h0,
                                   const float* __restrict__ h1) {
    int i = blockIdx.x * blockDim.x + threadIdx.x;
    if (i < BB * HH) {
        dst[i] = h0[i];
        dst[BB * HH + i] = h1[i];
    }
}

extern "C" void kernel_launch(void* const* d_in, const int* in_sizes, int n_in,
                              void* d_out, int out_size, void* d_ws, size_t ws_size,
                              hipStream_t stream) {
    (void)in_sizes; (void)n_in; (void)out_size; (void)ws_size;
    const float* x = (const float*)d_in[0];
    // params flattened: for (layer, dir): w_ih, w_hh, b_ih, b_hh
    const float* W[2][2][4];
    for (int l = 0; l < 2; ++l)
        for (int d = 0; d < 2; ++d)
            for (int k = 0; k < 4; ++k)
                W[l][d][k] = (const float*)d_in[1 + (l * 2 + d) * 4 + k];

    float* out = (float*)d_out;
    float* y_final = out;                                      // (B,S,2H)
    float* hid_out = out + (size_t)BB * SS * (2 * HH);         // (4,B,H)

    float* ws = (float*)d_ws;
    float* y0 = ws;                                            // (B,S,2H) layer-0 output
    float* h0bufs = y0 + (size_t)BB * SS * (2 * HH);           // 2*B*H
    float* h1bufs = h0bufs + 2 * BB * HH;                      // 2*B*H

    const dim3 grid(HH / 16, 2);
    const dim3 block(128);
    const int hz = 4 * BB * HH;  // zero both dirs' double buffers (contiguous)

    // ---------------- Layer 0 (input = x, IN_SZ = 256) ----------------
    zero_f32_kernel<<<(hz + 255) / 256, 256, 0, stream>>>(h0bufs, hz);
    for (int s = 0; s < SS; ++s) {
        gru_step_kernel<II><<<grid, block, 0, stream>>>(
            x,
            W[0][0][0], W[0][0][1], W[0][0][2], W[0][0][3],
            W[0][1][0], W[0][1][1], W[0][1][2], W[0][1][3],
            h0bufs, h1bufs, y0, s);
    }
    // final h lands in buffer index (SS % 2) == 0
    copy_hidden_kernel<<<(BB * HH + 255) / 256, 256, 0, stream>>>(hid_out, h0bufs, h1bufs);

    // ---------------- Layer 1 (input = y0, IN_SZ = 1024) ----------------
    zero_f32_kernel<<<(hz + 255) / 256, 256, 0, stream>>>(h0bufs, hz);
    for (int s = 0; s < SS; ++s) {
        gru_step_kernel<2 * HH><<<grid, block, 0, stream>>>(
            y0,
            W[1][0][0], W[1][0][1], W[1][0][2], W[1][0][3],
            W[1][1][0], W[1][1][1], W[1][1][2], W[1][1][3],
            h0bufs, h1bufs, y_final, s);
    }
    copy_hidden_kernel<<<(BB * HH + 255) / 256, 256, 0, stream>>>(hid_out + 2 * BB * HH, h0bufs, h1bufs);
}